// NetFV_25666724561114
// MI455X (gfx1250) — compile-verified
//
#include <hip/hip_runtime.h>
#include <hip/hip_bf16.h>

typedef __attribute__((ext_vector_type(2))) float v2f;
typedef __attribute__((ext_vector_type(4))) float v4f;
typedef __attribute__((ext_vector_type(8))) float v8f;

#define D_    1024
#define K_    64
#define N_    512
#define B_    32
#define OUT_  1024
#define BN_   (B_ * N_)      // 16384 rows
#define DK_   (D_ * K_)      // 65536
#define SPLIT_ 128           // split-K factor for the big GEMM

__device__ __forceinline__ v8f wmma4(v2f a, v2f b, v8f c) {
  // D = A(16x4,f32) * B(4x16,f32) + C(16x16,f32)
  return __builtin_amdgcn_wmma_f32_16x16x4_f32(false, a, false, b, (short)0, c,
                                               false, false);
}

__device__ __forceinline__ v8f zero8() {
  v8f c;
#pragma unroll
  for (int i = 0; i < 8; ++i) c[i] = 0.f;
  return c;
}

// ---------------------------------------------------------------------------
// K1: act[16384,64] = x[16384,1024] @ W[1024,64]   (WMMA f32 16x16x4)
// One wave per 16x16 output tile; 8 waves/block.
// ---------------------------------------------------------------------------
__global__ __launch_bounds__(256) void k1_act_gemm(const float* __restrict__ x,
                                                   const float* __restrict__ W,
                                                   float* __restrict__ act) {
  const int tid = threadIdx.x;
  const int lane = tid & 31, wave = tid >> 5;
  const int half = lane >> 4, ln = lane & 15;
  const int g = blockIdx.x * 8 + wave;
  const int mtile = g >> 2, ntile = g & 3;

  const float* arow = x + (size_t)(mtile * 16 + ln) * D_;
  v8f c = zero8();
#pragma unroll 4
  for (int k0 = 0; k0 < D_; k0 += 4) {
    v2f a = *(const v2f*)(arow + k0 + 2 * half);
    v2f b;
    b.x = W[(size_t)(k0 + 2 * half) * K_ + ntile * 16 + ln];
    b.y = W[(size_t)(k0 + 2 * half + 1) * K_ + ntile * 16 + ln];
    c = wmma4(a, b, c);
  }
#pragma unroll
  for (int r = 0; r < 8; ++r) {
    int m = r + 8 * half;
    act[(size_t)(mtile * 16 + m) * K_ + ntile * 16 + ln] = c[r];
  }
}

// ---------------------------------------------------------------------------
// K2a/K2b: deterministic BN1 stats over 16384 rows (per column of 64)
// ---------------------------------------------------------------------------
__global__ __launch_bounds__(256) void k2a_bnpart(const float* __restrict__ act,
                                                  float* __restrict__ partS,
                                                  float* __restrict__ partQ) {
  __shared__ float rs[256], rq[256];
  const int t = threadIdx.x, k = t & 63, gsub = t >> 6;
  const int c = blockIdx.x;
  float s = 0.f, q = 0.f;
  const int rowBase = c * 256 + gsub * 64;
  for (int j = 0; j < 64; ++j) {
    float v = act[(size_t)(rowBase + j) * K_ + k];
    s += v;
    q += v * v;
  }
  rs[t] = s;
  rq[t] = q;
  __syncthreads();
  if (t < 64) {
    s = rs[k] + rs[64 + k] + rs[128 + k] + rs[192 + k];
    q = rq[k] + rq[64 + k] + rq[128 + k] + rq[192 + k];
    partS[c * 64 + k] = s;
    partQ[c * 64 + k] = q;
  }
}

__global__ __launch_bounds__(64) void k2b_bnfin(const float* __restrict__ partS,
                                                const float* __restrict__ partQ,
                                                const float* __restrict__ gamma,
                                                const float* __restrict__ beta,
                                                float* __restrict__ bscale,
                                                float* __restrict__ bshift) {
  const int k = threadIdx.x;
  float s = 0.f, q = 0.f;
  for (int c = 0; c < 64; ++c) {
    s += partS[c * 64 + k];
    q += partQ[c * 64 + k];
  }
  float mean = s * (1.f / (float)BN_);
  float var = q * (1.f / (float)BN_) - mean * mean;
  float istd = rsqrtf(var + 1e-5f);
  float sc = gamma[k] * istd;
  bscale[k] = sc;
  bshift[k] = beta[k] - mean * sc;
}

// ---------------------------------------------------------------------------
// K3: BN affine + row softmax over K=64, one wave per row, in place.
// ---------------------------------------------------------------------------
__global__ __launch_bounds__(256) void k3_softmax(float* __restrict__ act,
                                                  const float* __restrict__ bscale,
                                                  const float* __restrict__ bshift) {
  const int tid = threadIdx.x;
  const int lane = tid & 31, wave = tid >> 5;
  const int row = blockIdx.x * 8 + wave;
  float* rp = act + (size_t)row * K_;
  float v0 = rp[lane] * bscale[lane] + bshift[lane];
  float v1 = rp[lane + 32] * bscale[lane + 32] + bshift[lane + 32];
  float m = fmaxf(v0, v1);
#pragma unroll
  for (int off = 16; off >= 1; off >>= 1) m = fmaxf(m, __shfl_xor(m, off, 32));
  float e0 = expf(v0 - m), e1 = expf(v1 - m);
  float s = e0 + e1;
#pragma unroll
  for (int off = 16; off >= 1; off >>= 1) s += __shfl_xor(s, off, 32);
  float inv = 1.f / s;
  rp[lane] = e0 * inv;
  rp[lane + 32] = e1 * inv;
}

// ---------------------------------------------------------------------------
// K4: a_sum[b,k] = sum_n act[b,n,k]   (deterministic)
// ---------------------------------------------------------------------------
__global__ __launch_bounds__(64) void k4_asum(const float* __restrict__ act,
                                              float* __restrict__ asum) {
  const int b = blockIdx.x, k = threadIdx.x;
  float s = 0.f;
  const float* p = act + (size_t)b * N_ * K_ + k;
  for (int n = 0; n < N_; ++n) s += p[(size_t)n * K_];
  asum[b * K_ + k] = s;
}

// ---------------------------------------------------------------------------
// K5: fused einsums fv1[b,d,k]=sum_n act*x, fv2raw=sum_n act*x^2 (WMMA)
// Block: (b, 64-wide d-tile). LDS-staged x and act chunks, x^2 in-register.
// Epilogue applies the Fisher-vector elementwise math.
// ---------------------------------------------------------------------------
__global__ __launch_bounds__(256) void k5_fv_gemm(
    const float* __restrict__ x, const float* __restrict__ act,
    const float* __restrict__ asum, const float* __restrict__ cw2w,
    const float* __restrict__ covw, float* __restrict__ fv1,
    float* __restrict__ fv2) {
  __shared__ float sx[16][64];
  __shared__ float sa[16][64];
  const int tid = threadIdx.x;
  const int lane = tid & 31, wave = tid >> 5;
  const int half = lane >> 4, ln = lane & 15;
  const int b = blockIdx.x >> 4;
  const int dBase = (blockIdx.x & 15) * 64;
  const int p0 = wave * 2;
  const int dt = p0 >> 2, kt0 = p0 & 3, kt1 = kt0 + 1;
  const int li = tid >> 4, lc = (tid & 15) * 4;

  v8f acc10 = zero8(), acc11 = zero8(), acc20 = zero8(), acc21 = zero8();

  for (int n0 = 0; n0 < N_; n0 += 16) {
    v4f vx = *(const v4f*)(x + ((size_t)(b * N_ + n0 + li)) * D_ + dBase + lc);
    v4f va = *(const v4f*)(act + ((size_t)(b * N_ + n0 + li)) * K_ + lc);
    __syncthreads();
    *(v4f*)&sx[li][lc] = vx;
    *(v4f*)&sa[li][lc] = va;
    __syncthreads();
#pragma unroll
    for (int kk = 0; kk < 16; kk += 4) {
      int kr = kk + 2 * half;
      v2f a;
      a.x = sx[kr][dt * 16 + ln];
      a.y = sx[kr + 1][dt * 16 + ln];
      v2f a2 = a * a;
      v2f b0, b1;
      b0.x = sa[kr][kt0 * 16 + ln];
      b0.y = sa[kr + 1][kt0 * 16 + ln];
      b1.x = sa[kr][kt1 * 16 + ln];
      b1.y = sa[kr + 1][kt1 * 16 + ln];
      acc10 = wmma4(a, b0, acc10);
      acc20 = wmma4(a2, b0, acc20);
      acc11 = wmma4(a, b1, acc11);
      acc21 = wmma4(a2, b1, acc21);
    }
  }

#pragma unroll
  for (int j = 0; j < 2; ++j) {
    int kt = (j == 0) ? kt0 : kt1;
    v8f f1v = (j == 0) ? acc10 : acc11;
    v8f f2v = (j == 0) ? acc20 : acc21;
#pragma unroll
    for (int r = 0; r < 8; ++r) {
      int d = dBase + dt * 16 + r + 8 * half;
      int k = kt * 16 + ln;
      size_t dk = (size_t)d * K_ + k;
      float c2 = cw2w[dk];
      float cv = covw[dk];
      float cw = cv * cv + 1e-6f;
      float as = asum[b * K_ + k];
      float f1 = f1v[r], f2 = f2v[r];
      float o1 = (f1 - as * c2) / cw;
      float o2 = (as * c2 * c2 + f2 - 2.f * f1 * c2) / (cw * cw) - as;
      size_t idx = (size_t)b * DK_ + dk;
      fv1[idx] = o1;
      fv2[idx] = o2;
    }
  }
}

// ---------------------------------------------------------------------------
// K6: per-batch L2-norm factors.  fv1: per-(b,k) column norm over D then
// whole-vector norm.  fv2: whole-vector norm (applied twice as in ref).
// ---------------------------------------------------------------------------
__global__ __launch_bounds__(256) void k6_norm(const float* __restrict__ fv1,
                                               const float* __restrict__ fv2,
                                               float* __restrict__ scale1,
                                               float* __restrict__ scale2) {
  __shared__ float r1[256], r2[256];
  __shared__ float shf1;
  const int t = threadIdx.x, k = t & 63, gsub = t >> 6;
  const int b = blockIdx.x;
  float s1 = 0.f, s2 = 0.f;
  const float* p1 = fv1 + (size_t)b * DK_ + k;
  const float* p2 = fv2 + (size_t)b * DK_ + k;
  for (int dd = 0; dd < 256; ++dd) {
    int d = gsub * 256 + dd;
    float a = p1[(size_t)d * K_];
    float c = p2[(size_t)d * K_];
    s1 += a * a;
    s2 += c * c;
  }
  r1[t] = s1;
  r2[t] = s2;
  __syncthreads();
  float ss1 = 0.f, ss2 = 0.f;
  if (t < 64) {
    ss1 = r1[k] + r1[64 + k] + r1[128 + k] + r1[192 + k];
    ss2 = r2[k] + r2[64 + k] + r2[128 + k] + r2[192 + k];
    float n1 = sqrtf(ss1);
    float d1 = fmaxf(n1, 1e-12f);
    scale1[b * 64 + k] = 1.f / d1;  // per-column factor (provisional)
  }
  __syncthreads();
  if (t < 64) {
    float n1 = sqrtf(ss1);
    float d1 = fmaxf(n1, 1e-12f);
    r1[k] = ss1 / (d1 * d1);  // sumsq after column normalization
    r2[k] = ss2;
  }
  __syncthreads();
  if (t == 0) {
    float t1 = 0.f, t2 = 0.f;
    for (int i = 0; i < 64; ++i) {
      t1 += r1[i];
      t2 += r2[i];
    }
    float nt1 = sqrtf(t1);
    shf1 = 1.f / fmaxf(nt1, 1e-12f);
    float na = sqrtf(t2);
    float sa = 1.f / fmaxf(na, 1e-12f);
    float nb = na * sa;  // norm after first normalization
    float sb = 1.f / fmaxf(nb, 1e-12f);
    scale2[b] = sa * sb;
  }
  __syncthreads();
  if (t < 64) scale1[b * 64 + k] *= shf1;
}

// ---------------------------------------------------------------------------
// K7: apply normalization scales in place.
// ---------------------------------------------------------------------------
__global__ __launch_bounds__(256) void k7_scale(float* __restrict__ fv1,
                                                float* __restrict__ fv2,
                                                const float* __restrict__ scale1,
                                                const float* __restrict__ scale2) {
  size_t idx = (size_t)blockIdx.x * 256 + threadIdx.x;
  int b = (int)(idx >> 16);
  int k = (int)(idx & 63);
  fv1[idx] *= scale1[b * 64 + k];
  fv2[idx] *= scale2[b];
}

// ---------------------------------------------------------------------------
// K8: big GEMM  out[32,1024] = fvcat[32,131072] @ H[131072,1024]
// Split-K (128 chunks of 1024) x 8 N-blocks = 1024 blocks to saturate HBM.
// Each wave: 32x16 tile (two WMMA per k-step share the B fragment so H is
// streamed exactly once).  A chunk staged in padded LDS.
// ---------------------------------------------------------------------------
__global__ __launch_bounds__(256) void k8_big_gemm(const float* __restrict__ fv1,
                                                   const float* __restrict__ fv2,
                                                   const float* __restrict__ H,
                                                   float* __restrict__ partial) {
  __shared__ float As[32][132];
  const int tid = threadIdx.x;
  const int lane = tid & 31, wave = tid >> 5;
  const int half = lane >> 4, ln = lane & 15;
  const int kc = blockIdx.x >> 3;
  const int nb = blockIdx.x & 7;
  const int kbase = kc * 1024;
  const float* Abuf = (kbase < DK_) ? fv1 : fv2;
  const int koff = kbase & (DK_ - 1);
  const int ncolg = nb * 128 + wave * 16 + ln;
  const int lr = tid >> 3, lc = (tid & 7) * 16;

  v8f acc0 = zero8(), acc1 = zero8();

  for (int ck = 0; ck < 8; ++ck) {
    const float* src = Abuf + (size_t)lr * DK_ + koff + ck * 128 + lc;
    v4f t0 = *(const v4f*)(src);
    v4f t1 = *(const v4f*)(src + 4);
    v4f t2 = *(const v4f*)(src + 8);
    v4f t3 = *(const v4f*)(src + 12);
    __syncthreads();
    *(v4f*)&As[lr][lc] = t0;
    *(v4f*)&As[lr][lc + 4] = t1;
    *(v4f*)&As[lr][lc + 8] = t2;
    *(v4f*)&As[lr][lc + 12] = t3;
    __syncthreads();
    const float* Hp = H + (size_t)(kbase + ck * 128) * OUT_ + ncolg;
#pragma unroll 4
    for (int kk = 0; kk < 128; kk += 4) {
      int kr = kk + 2 * half;
      v2f a0 = *(const v2f*)&As[ln][kr];
      v2f a1 = *(const v2f*)&As[16 + ln][kr];
      v2f b;
      b.x = Hp[(size_t)kr * OUT_];
      b.y = Hp[(size_t)(kr + 1) * OUT_];
      acc0 = wmma4(a0, b, acc0);
      acc1 = wmma4(a1, b, acc1);
    }
  }
#pragma unroll
  for (int r = 0; r < 8; ++r) {
    int m = r + 8 * half;
    partial[((size_t)kc * 32 + m) * OUT_ + ncolg] = acc0[r];
    partial[((size_t)kc * 32 + 16 + m) * OUT_ + ncolg] = acc1[r];
  }
}

// ---------------------------------------------------------------------------
// K9: deterministic split-K reduction.
// ---------------------------------------------------------------------------
__global__ __launch_bounds__(256) void k9_reduce(const float* __restrict__ partial,
                                                 float* __restrict__ fvout) {
  const int o = blockIdx.x * 256 + threadIdx.x;
  float s = 0.f;
  for (int c = 0; c < SPLIT_; ++c) s += partial[(size_t)c * (32 * OUT_) + o];
  fvout[o] = s;
}

// ---------------------------------------------------------------------------
// K10: gating GEMM  gates[32,1024] = fvout @ G[1024,1024]  (WMMA, small)
// ---------------------------------------------------------------------------
__global__ __launch_bounds__(256) void k10_gate_gemm(const float* __restrict__ fvout,
                                                     const float* __restrict__ G,
                                                     float* __restrict__ gates) {
  const int tid = threadIdx.x;
  const int lane = tid & 31, wave = tid >> 5;
  const int half = lane >> 4, ln = lane & 15;
  const int ncolg = blockIdx.x * 128 + wave * 16 + ln;
  v8f acc0 = zero8(), acc1 = zero8();
#pragma unroll 4
  for (int k0 = 0; k0 < OUT_; k0 += 4) {
    v2f a0 = *(const v2f*)(fvout + (size_t)ln * OUT_ + k0 + 2 * half);
    v2f a1 = *(const v2f*)(fvout + (size_t)(16 + ln) * OUT_ + k0 + 2 * half);
    v2f b;
    b.x = G[(size_t)(k0 + 2 * half) * OUT_ + ncolg];
    b.y = G[(size_t)(k0 + 2 * half + 1) * OUT_ + ncolg];
    acc0 = wmma4(a0, b, acc0);
    acc1 = wmma4(a1, b, acc1);
  }
#pragma unroll
  for (int r = 0; r < 8; ++r) {
    int m = r + 8 * half;
    gates[(size_t)m * OUT_ + ncolg] = acc0[r];
    gates[(size_t)(16 + m) * OUT_ + ncolg] = acc1[r];
  }
}

// ---------------------------------------------------------------------------
// K11: BN2 (batch stats over 32 rows) + sigmoid + final gating product.
// ---------------------------------------------------------------------------
__global__ __launch_bounds__(256) void k11_bn2_out(const float* __restrict__ gates,
                                                   const float* __restrict__ fvout,
                                                   const float* __restrict__ gamma,
                                                   const float* __restrict__ beta,
                                                   float* __restrict__ out) {
  const int o = blockIdx.x * 256 + threadIdx.x;
  float s = 0.f, q = 0.f;
  for (int b = 0; b < B_; ++b) {
    float v = gates[(size_t)b * OUT_ + o];
    s += v;
    q += v * v;
  }
  float mean = s * (1.f / (float)B_);
  float var = q * (1.f / (float)B_) - mean * mean;
  float istd = rsqrtf(var + 1e-5f);
  float sc = gamma[o] * istd;
  float sh = beta[o] - mean * sc;
  for (int b = 0; b < B_; ++b) {
    float g = gates[(size_t)b * OUT_ + o] * sc + sh;
    float sig = 1.f / (1.f + expf(-g));
    out[(size_t)b * OUT_ + o] = fvout[(size_t)b * OUT_ + o] * sig;
  }
}

// ---------------------------------------------------------------------------
extern "C" void kernel_launch(void* const* d_in, const int* in_sizes, int n_in,
                              void* d_out, int out_size, void* d_ws, size_t ws_size,
                              hipStream_t stream) {
  (void)in_sizes;
  (void)n_in;
  (void)out_size;
  (void)ws_size;
  const float* x = (const float*)d_in[0];
  const float* cw = (const float*)d_in[1];    // cluster_weights [D,K]
  const float* covw = (const float*)d_in[2];  // covar_weights   [D,K]
  const float* cw2 = (const float*)d_in[3];   // cluster_weights2 [1,D,K]
  const float* H = (const float*)d_in[4];     // hidden1_weights [2DK,OUT]
  const float* g1 = (const float*)d_in[5];
  const float* b1 = (const float*)d_in[6];
  const float* G = (const float*)d_in[7];     // gating_weights [OUT,OUT]
  const float* g2 = (const float*)d_in[8];
  const float* b2 = (const float*)d_in[9];
  float* out = (float*)d_out;

  float* ws = (float*)d_ws;
  size_t off = 0;
  float* act = ws + off;    off += (size_t)BN_ * K_;        // 4 MB
  float* partS = ws + off;  off += 64 * 64;
  float* partQ = ws + off;  off += 64 * 64;
  float* bscale = ws + off; off += 64;
  float* bshift = ws + off; off += 64;
  float* asum = ws + off;   off += B_ * K_;
  float* fv1 = ws + off;    off += (size_t)B_ * DK_;        // 8 MB
  float* fv2 = ws + off;    off += (size_t)B_ * DK_;        // 8 MB
  float* scale1 = ws + off; off += B_ * K_;
  float* scale2 = ws + off; off += B_;
  float* partial = ws + off; off += (size_t)SPLIT_ * 32 * OUT_;  // 16 MB
  float* fvout = ws + off;  off += B_ * OUT_;
  float* gates = ws + off;  off += B_ * OUT_;

  k1_act_gemm<<<512, 256, 0, stream>>>(x, cw, act);
  k2a_bnpart<<<64, 256, 0, stream>>>(act, partS, partQ);
  k2b_bnfin<<<1, 64, 0, stream>>>(partS, partQ, g1, b1, bscale, bshift);
  k3_softmax<<<2048, 256, 0, stream>>>(act, bscale, bshift);
  k4_asum<<<32, 64, 0, stream>>>(act, asum);
  k5_fv_gemm<<<512, 256, 0, stream>>>(x, act, asum, cw2, covw, fv1, fv2);
  k6_norm<<<32, 256, 0, stream>>>(fv1, fv2, scale1, scale2);
  k7_scale<<<8192, 256, 0, stream>>>(fv1, fv2, scale1, scale2);
  k8_big_gemm<<<1024, 256, 0, stream>>>(fv1, fv2, H, partial);
  k9_reduce<<<128, 256, 0, stream>>>(partial, fvout);
  k10_gate_gemm<<<8, 256, 0, stream>>>(fvout, G, gates);
  k11_bn2_out<<<4, 256, 0, stream>>>(gates, fvout, g2, b2, out);
}